// TrimmedConv_3178275799592
// MI455X (gfx1250) — compile-verified
//
#include <hip/hip_runtime.h>

// Problem constants (match reference)
#define N_NODES 50000
#define KNBR    32
#define DIM     256     // D_IN == D_OUT == 256
#define REMOVE  14      // floor(32 * 0.45)
#define KEEP    4       // 32 - 2*14

typedef __attribute__((ext_vector_type(2))) float v2f;
typedef __attribute__((ext_vector_type(8))) float v8f;

// ---------------------------------------------------------------------------
// Kernel 1: h = x @ W^T via V_WMMA_F32_16X16X4_F32 (CDNA5 fp32 matrix op)
//   grid  = N/16 = 3125 blocks, block = 128 threads = 4 waves (wave32)
//   each wave owns 4 adjacent 16x16 output tiles (64 output columns)
// A (16x4, f32, 2 VGPRs/lane): lanes 0-15 -> K={0,1}, lanes 16-31 -> K={2,3}
// B (4x16, f32, 2 VGPRs/lane): same K split, lane&15 selects output column
// C/D (16x16 f32, 8 VGPRs): VGPR j -> M=j (lanes 0-15) / M=j+8 (lanes 16-31)
// ---------------------------------------------------------------------------
__global__ __launch_bounds__(128) void gemm_wmma_f32(
    const float* __restrict__ x,   // (N, 256) row-major
    const float* __restrict__ W,   // (256, 256) row-major (D_OUT, D_IN)
    float* __restrict__ h)         // (N, 256) row-major
{
    const int wave   = threadIdx.x >> 5;     // 0..3
    const int lane   = threadIdx.x & 31;
    const int n_base = blockIdx.x * 16;
    const int lrow   = lane & 15;            // M row / N col / O row within tile
    const int kc     = (lane >> 4) << 1;     // 0 for lanes 0-15, 2 for lanes 16-31
    const int o0     = wave * 64;            // this wave's first output column

    const float* xrow = x + (size_t)(n_base + lrow) * DIM;
    const float* w0   = W + (size_t)(o0 +  0 + lrow) * DIM;
    const float* w1   = W + (size_t)(o0 + 16 + lrow) * DIM;
    const float* w2   = W + (size_t)(o0 + 32 + lrow) * DIM;
    const float* w3   = W + (size_t)(o0 + 48 + lrow) * DIM;

    v8f acc0 = {}, acc1 = {}, acc2 = {}, acc3 = {};

    #pragma unroll 4
    for (int kb = 0; kb < DIM; kb += 4) {
        v2f a  = *(const v2f*)(xrow + kb + kc);
        v2f b0 = *(const v2f*)(w0   + kb + kc);
        v2f b1 = *(const v2f*)(w1   + kb + kc);
        v2f b2 = *(const v2f*)(w2   + kb + kc);
        v2f b3 = *(const v2f*)(w3   + kb + kc);
        // 8-arg form: (neg_a, A, neg_b, B, c_mod, C, reuse_a, reuse_b)
        acc0 = __builtin_amdgcn_wmma_f32_16x16x4_f32(false, a, false, b0,
                                                     (short)0, acc0, false, false);
        acc1 = __builtin_amdgcn_wmma_f32_16x16x4_f32(false, a, false, b1,
                                                     (short)0, acc1, false, false);
        acc2 = __builtin_amdgcn_wmma_f32_16x16x4_f32(false, a, false, b2,
                                                     (short)0, acc2, false, false);
        acc3 = __builtin_amdgcn_wmma_f32_16x16x4_f32(false, a, false, b3,
                                                     (short)0, acc3, false, false);
    }

    // Scatter C/D fragment back: VGPR j holds M=j (lanes 0-15) or M=j+8 (16-31)
    const int col = lane & 15;
    const int mhi = (lane >> 4) * 8;
    #pragma unroll
    for (int j = 0; j < 8; ++j) {
        const size_t row = (size_t)(n_base + mhi + j) * DIM;
        h[row + o0 +  0 + col] = acc0[j];
        h[row + o0 + 16 + col] = acc1[j];
        h[row + o0 + 32 + col] = acc2[j];
        h[row + o0 + 48 + col] = acc3[j];
    }
}

// ---------------------------------------------------------------------------
// Kernel 2: gather h[nbrs[n][k]][d] for k=0..31 (L2-resident: h is 51MB of
// 192MB L2), bitonic-sort the 32 values in registers (240 compare-exchanges,
// fully unrolled), emit mean of ranks 14..17.
//   grid = N blocks, block = 256 threads (thread = channel d)
// ---------------------------------------------------------------------------
__global__ __launch_bounds__(256) void trimmed_mean(
    const float* __restrict__ h,     // (N, 256)
    const int*   __restrict__ nbrs,  // (N, 32)
    float*       __restrict__ out)   // (N, 256)
{
    __shared__ int snbr[KNBR];
    const int n = blockIdx.x;
    const int d = threadIdx.x;

    if (d < KNBR) snbr[d] = nbrs[(size_t)n * KNBR + d];
    __syncthreads();

    // 32 independent coalesced row reads -> 32 outstanding LOADcnt slots,
    // latency fully overlapped, all hits in global L2.
    float v[KNBR];
    #pragma unroll
    for (int k = 0; k < KNBR; ++k) {
        v[k] = h[(size_t)snbr[k] * DIM + d];
    }

    // Bitonic sort (ascending) of 32 register values.
    #pragma unroll
    for (int kk = 2; kk <= KNBR; kk <<= 1) {
        #pragma unroll
        for (int j = kk >> 1; j > 0; j >>= 1) {
            #pragma unroll
            for (int i = 0; i < KNBR; ++i) {
                const int l = i ^ j;
                if (l > i) {
                    const bool up = ((i & kk) == 0);
                    const float lo = fminf(v[i], v[l]);
                    const float hi = fmaxf(v[i], v[l]);
                    v[i] = up ? lo : hi;
                    v[l] = up ? hi : lo;
                }
            }
        }
    }

    float s = 0.0f;
    #pragma unroll
    for (int r = REMOVE; r < REMOVE + KEEP; ++r) s += v[r];
    out[(size_t)n * DIM + d] = s * (1.0f / KEEP);
}

// ---------------------------------------------------------------------------
extern "C" void kernel_launch(void* const* d_in, const int* in_sizes, int n_in,
                              void* d_out, int out_size, void* d_ws, size_t ws_size,
                              hipStream_t stream) {
    const float* x    = (const float*)d_in[0];   // (N, 256) fp32
    const int*   nbrs = (const int*)  d_in[1];   // (N, 32)  int
    const float* W    = (const float*)d_in[2];   // (256, 256) fp32
    float*       out  = (float*)d_out;           // (N, 256) fp32
    float*       h    = (float*)d_ws;            // (N, 256) fp32 scratch = 51.2 MB

    (void)in_sizes; (void)n_in; (void)out_size; (void)ws_size;

    dim3 gemm_grid(N_NODES / 16);                // 3125, exact (50000 = 16*3125)
    dim3 gemm_block(128);                        // 4 waves, each 4 o-tiles
    gemm_wmma_f32<<<gemm_grid, gemm_block, 0, stream>>>(x, W, h);

    dim3 trim_grid(N_NODES);
    dim3 trim_block(DIM);                        // 256 threads = 8 waves
    trimmed_mean<<<trim_grid, trim_block, 0, stream>>>(h, nbrs, out);
}